// AttentionDecoder_18691697672831
// MI455X (gfx1250) — compile-verified
//
#include <hip/hip_runtime.h>
#include <math.h>

// ---------------------------------------------------------------------------
// AttentionDecoder for MI455X (gfx1250, wave32).
// All GEMMs run on V_WMMA_F32_16X16X4_F32 (full fp32, matches reference).
// M-block register blocking: each wave computes MB 16x16 C-tiles sharing one
// W fragment per k-step (cuts L2 weight traffic MB-fold on the dominant
// vocabulary GEMM, whose 20.5 MB weight matrix lives in the 192 MB L2).
// ---------------------------------------------------------------------------

typedef __attribute__((ext_vector_type(2))) float v2f;
typedef __attribute__((ext_vector_type(8))) float v8f;

#define B_   128
#define T_   20
#define TS_  19          // T-1 decode steps
#define R_   49
#define V_   10000
#define E_   512
#define H_   512
#define FD_  512
#define A_   256

__device__ __forceinline__ v8f wmma4(v2f a, v2f w, v8f c) {
    return __builtin_amdgcn_wmma_f32_16x16x4_f32(
        false, a, false, w, (short)0, c, false, false);
}

// ---------------------------------------------------------------------------
// Plain TN GEMM (one 16x16 tile per wave): C[m,n] = sum_k A[m,k]W[n,k]+bias[n]
// Used for the tiny latency-bound h-projection GEMM.
// ---------------------------------------------------------------------------
__global__ void gemm_tn_bias(const float* __restrict__ A, int lda,
                             const float* __restrict__ W, int ldw,
                             const float* __restrict__ bias,
                             float* __restrict__ C, int ldc,
                             int Mtiles, int Ntiles, int K)
{
    const int wave = (blockIdx.x * blockDim.x + threadIdx.x) >> 5;
    const int lane = threadIdx.x & 31;
    if (wave >= Mtiles * Ntiles) return;

    const int mt   = wave / Ntiles;
    const int nt   = wave % Ntiles;
    const int half = lane >> 4;      // 0: K pair {0,1}, 1: K pair {2,3}
    const int row  = lane & 15;

    const float* a = A + (size_t)(mt * 16 + row) * lda + 2 * half;
    const float* w = W + (size_t)(nt * 16 + row) * ldw + 2 * half;

    v8f acc = {};
    for (int k = 0; k < K; k += 4)
        acc = wmma4(*(const v2f*)(a + k), *(const v2f*)(w + k), acc);

    const int   n  = nt * 16 + row;
    const float bv = bias ? bias[n] : 0.0f;
    const int   mb = mt * 16 + 8 * half;
#pragma unroll
    for (int r = 0; r < 8; ++r)
        C[(size_t)(mb + r) * ldc + n] = acc[r] + bv;
}

// ---------------------------------------------------------------------------
// M-blocked TN GEMM: each wave computes MB consecutive M-tiles for one N-tile.
// The W fragment is loaded once per k-step and reused by MB WMMAs.
// Requires (Mtiles % MB)==0.
// ---------------------------------------------------------------------------
template<int MB>
__global__ void gemm_tn_bias_mblk(const float* __restrict__ A, int lda,
                                  const float* __restrict__ W, int ldw,
                                  const float* __restrict__ bias,
                                  float* __restrict__ C, int ldc,
                                  int Mblocks, int Ntiles, int K)
{
    const int wave = (blockIdx.x * blockDim.x + threadIdx.x) >> 5;
    const int lane = threadIdx.x & 31;
    if (wave >= Mblocks * Ntiles) return;

    const int mb   = wave / Ntiles;
    const int nt   = wave % Ntiles;
    const int half = lane >> 4;
    const int row  = lane & 15;

    const float* w = W + (size_t)(nt * 16 + row) * ldw + 2 * half;
    const float* a = A + (size_t)(mb * MB * 16 + row) * lda + 2 * half;

    v8f acc[MB];
#pragma unroll
    for (int i = 0; i < MB; ++i) acc[i] = (v8f){};

    for (int k = 0; k < K; k += 4) {
        v2f wv = *(const v2f*)(w + k);
#pragma unroll
        for (int i = 0; i < MB; ++i) {
            v2f av = *(const v2f*)(a + (size_t)i * 16 * lda + k);
            acc[i] = wmma4(av, wv, acc[i]);
        }
    }

    const int   n  = nt * 16 + row;
    const float bv = bias ? bias[n] : 0.0f;
#pragma unroll
    for (int i = 0; i < MB; ++i) {
        const int mrow = mb * MB * 16 + i * 16 + 8 * half;
#pragma unroll
        for (int r = 0; r < 8; ++r)
            C[(size_t)(mrow + r) * ldc + n] = acc[i][r] + bv;
    }
}

// ---------------------------------------------------------------------------
// Gates GEMM (M-blocked by 4): gates[b,j] = [x_t|context|hx].[W_ih|W_hh](j,:)
//   + b_ih[j] + b_hh[j].  Three K segments of 512 into one accumulator set.
// ---------------------------------------------------------------------------
__global__ void gates_gemm(const float* __restrict__ xt, int ldx,  // embeds+t*E, ld=TS_*E_
                           const float* __restrict__ ctx,          // B x FD
                           const float* __restrict__ hx,           // B x H
                           const float* __restrict__ W_ih,         // (4H, E+FD)
                           const float* __restrict__ W_hh,         // (4H, H)
                           const float* __restrict__ b_ih,
                           const float* __restrict__ b_hh,
                           float* __restrict__ gates)              // B x 4H
{
    constexpr int MB = 4;
    const int Ntiles  = (4 * H_) / 16;            // 128
    const int Mblocks = B_ / (16 * MB);           // 2
    const int wave = (blockIdx.x * blockDim.x + threadIdx.x) >> 5;
    const int lane = threadIdx.x & 31;
    if (wave >= Mblocks * Ntiles) return;

    const int mb   = wave / Ntiles;
    const int nt   = wave % Ntiles;
    const int half = lane >> 4;
    const int row  = lane & 15;
    const int n    = nt * 16 + row;
    const int m0   = mb * MB * 16 + row;

    const float* w0 = W_ih + (size_t)n * (E_ + FD_) + 2 * half;
    const float* w1 = w0 + E_;
    const float* w2 = W_hh + (size_t)n * H_ + 2 * half;
    const float* a0 = xt  + (size_t)m0 * ldx + 2 * half;
    const float* a1 = ctx + (size_t)m0 * FD_ + 2 * half;
    const float* a2 = hx  + (size_t)m0 * H_  + 2 * half;

    v8f acc[MB];
#pragma unroll
    for (int i = 0; i < MB; ++i) acc[i] = (v8f){};

    for (int k = 0; k < E_; k += 4) {
        v2f wv = *(const v2f*)(w0 + k);
#pragma unroll
        for (int i = 0; i < MB; ++i)
            acc[i] = wmma4(*(const v2f*)(a0 + (size_t)i * 16 * ldx + k), wv, acc[i]);
    }
    for (int k = 0; k < FD_; k += 4) {
        v2f wv = *(const v2f*)(w1 + k);
#pragma unroll
        for (int i = 0; i < MB; ++i)
            acc[i] = wmma4(*(const v2f*)(a1 + (size_t)i * 16 * FD_ + k), wv, acc[i]);
    }
    for (int k = 0; k < H_; k += 4) {
        v2f wv = *(const v2f*)(w2 + k);
#pragma unroll
        for (int i = 0; i < MB; ++i)
            acc[i] = wmma4(*(const v2f*)(a2 + (size_t)i * 16 * H_ + k), wv, acc[i]);
    }

    const float bv = b_ih[n] + b_hh[n];
#pragma unroll
    for (int i = 0; i < MB; ++i) {
        const int mrow = mb * MB * 16 + i * 16 + 8 * half;
#pragma unroll
        for (int r = 0; r < 8; ++r)
            gates[(size_t)(mrow + r) * (4 * H_) + n] = acc[i][r] + bv;
    }
}

// ---------------------------------------------------------------------------
// Embedding gather: embeds[b][t][e] = emb[captions[b][t]][e], t in [0,19)
// ---------------------------------------------------------------------------
__global__ void embed_gather(const int* __restrict__ captions,
                             const float* __restrict__ emb,
                             float* __restrict__ embeds)
{
    int idx = blockIdx.x * blockDim.x + threadIdx.x;
    if (idx >= B_ * TS_ * E_) return;
    int e = idx % E_;
    int t = (idx / E_) % TS_;
    int b = idx / (E_ * TS_);
    int tok = captions[b * T_ + t];
    embeds[idx] = emb[(size_t)tok * E_ + e];
}

__global__ void zero_f32(float* __restrict__ p, int n)
{
    int idx = blockIdx.x * blockDim.x + threadIdx.x;
    if (idx < n) p[idx] = 0.0f;
}

// ---------------------------------------------------------------------------
// Attention epilogue (one block of 256 threads per batch element):
//   scores[r] = tanh(h[b,:] + f_proj[b,r,:]) . v_w + v_b
//   alpha     = softmax_r(scores)
//   context[b,f] = sum_r alpha[r] * features[b,r,f]
// ---------------------------------------------------------------------------
__global__ void attention_epilogue(const float* __restrict__ h,       // B x A
                                   const float* __restrict__ f_proj,  // B x R x A
                                   const float* __restrict__ v_w,     // A
                                   const float* __restrict__ v_b,     // scalar
                                   const float* __restrict__ features,// B x R x FD
                                   float* __restrict__ context)       // B x FD
{
    const int b    = blockIdx.x;
    const int tid  = threadIdx.x;
    const int lane = tid & 31;
    const int wid  = tid >> 5;                // 8 waves

    __shared__ float sc[R_];

    const float* hb = h + (size_t)b * A_;
    for (int r = wid; r < R_; r += 8) {
        const float* fp = f_proj + ((size_t)b * R_ + r) * A_;
        float s = 0.0f;
        for (int a = lane; a < A_; a += 32)
            s += tanhf(hb[a] + fp[a]) * v_w[a];
        for (int off = 16; off > 0; off >>= 1)
            s += __shfl_xor(s, off, 32);
        if (lane == 0) sc[r] = s + v_b[0];
    }
    __syncthreads();

    if (wid == 0) {                            // softmax over 49 scores
        float m = -3.0e38f;
        for (int r = lane; r < R_; r += 32) m = fmaxf(m, sc[r]);
        for (int off = 16; off > 0; off >>= 1) m = fmaxf(m, __shfl_xor(m, off, 32));
        float ssum = 0.0f;
        for (int r = lane; r < R_; r += 32) {
            float e = __expf(sc[r] - m);
            sc[r] = e;
            ssum += e;
        }
        for (int off = 16; off > 0; off >>= 1) ssum += __shfl_xor(ssum, off, 32);
        float inv = 1.0f / ssum;
        for (int r = lane; r < R_; r += 32) sc[r] *= inv;
    }
    __syncthreads();

    const float* fb = features + (size_t)b * R_ * FD_;
    for (int f = tid; f < FD_; f += blockDim.x) {
        float c = 0.0f;
        for (int r = 0; r < R_; ++r)
            c += sc[r] * fb[(size_t)r * FD_ + f];
        context[(size_t)b * FD_ + f] = c;
    }
}

// ---------------------------------------------------------------------------
// LSTM pointwise cell: gates order i,f,g,o; updates hx,cx in place.
// ---------------------------------------------------------------------------
__global__ void lstm_cell(const float* __restrict__ gates,
                          float* __restrict__ hx, float* __restrict__ cx)
{
    int idx = blockIdx.x * blockDim.x + threadIdx.x;
    if (idx >= B_ * H_) return;
    int b = idx / H_, hh = idx % H_;
    const float* g = gates + (size_t)b * (4 * H_);
    float gi = g[hh];
    float gf = g[H_ + hh];
    float gg = g[2 * H_ + hh];
    float go = g[3 * H_ + hh];
    float si = 1.0f / (1.0f + __expf(-gi));
    float sf = 1.0f / (1.0f + __expf(-gf));
    float so = 1.0f / (1.0f + __expf(-go));
    float c = sf * cx[idx] + si * tanhf(gg);
    cx[idx] = c;
    hx[idx] = so * tanhf(c);
}

// ---------------------------------------------------------------------------
// Host-side orchestration
// ---------------------------------------------------------------------------
extern "C" void kernel_launch(void* const* d_in, const int* in_sizes, int n_in,
                              void* d_out, int out_size, void* d_ws, size_t ws_size,
                              hipStream_t stream)
{
    const float* features = (const float*)d_in[0];
    const int*   captions = (const int*)  d_in[1];
    const float* emb      = (const float*)d_in[2];
    const float* W_h      = (const float*)d_in[3];
    const float* b_h      = (const float*)d_in[4];
    const float* W_f      = (const float*)d_in[5];
    const float* b_f      = (const float*)d_in[6];
    const float* v_w      = (const float*)d_in[7];
    const float* v_b      = (const float*)d_in[8];
    const float* W_ih     = (const float*)d_in[9];
    const float* W_hh     = (const float*)d_in[10];
    const float* b_ih     = (const float*)d_in[11];
    const float* b_hh     = (const float*)d_in[12];
    const float* fc_W     = (const float*)d_in[13];
    const float* fc_b     = (const float*)d_in[14];
    float*       out      = (float*)d_out;

    // workspace layout (floats)
    float* ws      = (float*)d_ws;
    float* embeds  = ws;                                    // B*TS*E
    float* f_proj  = embeds + (size_t)B_ * TS_ * E_;        // B*R*A
    float* h_buf   = f_proj + (size_t)B_ * R_ * A_;         // B*A
    float* context = h_buf  + (size_t)B_ * A_;              // B*FD
    float* gates   = context + (size_t)B_ * FD_;            // B*4H
    float* hx      = gates  + (size_t)B_ * 4 * H_;          // B*H
    float* cx      = hx     + (size_t)B_ * H_;              // B*H

    // 1) zero-init recurrent state (deterministic across graph replays)
    {
        int n = 2 * B_ * H_;
        zero_f32<<<(n + 255) / 256, 256, 0, stream>>>(hx, n);
    }
    // 2) embedding gather
    {
        int n = B_ * TS_ * E_;
        embed_gather<<<(n + 255) / 256, 256, 0, stream>>>(captions, emb, embeds);
    }
    // 3) f_proj = features @ W_f^T + b_f   (M=6272, N=256, K=512), MB=4
    {
        int Mblocks = (B_ * R_) / (16 * 4);   // 98
        int Ntiles  = A_ / 16;                // 16
        int waves   = Mblocks * Ntiles;       // 1568
        gemm_tn_bias_mblk<4><<<(waves + 3) / 4, 128, 0, stream>>>(
            features, FD_, W_f, FD_, b_f, f_proj, A_, Mblocks, Ntiles, FD_);
    }

    // 4) sequential decode
    for (int t = 0; t < TS_; ++t) {
        // h = hx @ W_h^T + b_h   (128 x 256, K=512): tiny, tile-per-wave
        {
            int Mtiles = B_ / 16, Ntiles = A_ / 16;   // 8 x 16 = 128 waves
            gemm_tn_bias<<<(Mtiles * Ntiles) / 4, 128, 0, stream>>>(
                hx, H_, W_h, H_, b_h, h_buf, A_, Mtiles, Ntiles, H_);
        }
        // attention: scores, softmax, context
        attention_epilogue<<<B_, 256, 0, stream>>>(
            h_buf, f_proj, v_w, v_b, features, context);

        // gates = [x_t|context] @ W_ih^T + hx @ W_hh^T + (b_ih+b_hh), MB=4
        {
            int waves = (B_ / (16 * 4)) * ((4 * H_) / 16);  // 2*128 = 256
            gates_gemm<<<(waves + 3) / 4, 128, 0, stream>>>(
                embeds + (size_t)t * E_, TS_ * E_, context, hx,
                W_ih, W_hh, b_ih, b_hh, gates);
        }
        // pointwise cell update
        {
            int n = B_ * H_;
            lstm_cell<<<(n + 255) / 256, 256, 0, stream>>>(gates, hx, cx);
        }
        // logits = hx @ fc_W^T + fc_b -> out[:, t, :]  (128 x 10000, K=512), MB=4
        {
            int Mblocks = B_ / (16 * 4);              // 2
            int Ntiles  = V_ / 16;                    // 625
            int waves   = Mblocks * Ntiles;           // 1250
            gemm_tn_bias_mblk<4><<<(waves + 3) / 4, 128, 0, stream>>>(
                hx, H_, fc_W, H_, fc_b, out + (size_t)t * V_, TS_ * V_,
                Mblocks, Ntiles, H_);
        }
    }
}